// MambaVideoEncoder_21706764714403
// MI455X (gfx1250) — compile-verified
//
#include <hip/hip_runtime.h>
#include <hip/hip_bf16.h>
#include <math.h>

// ---------------- problem constants ----------------
#define L     4096
#define DM    768
#define DI    1536          // D_INNER
#define DS    16            // D_STATE
#define RK    48            // DT_RANK
#define RB    64            // DT_RANK + D_STATE
#define XC    80            // W_xproj total cols
#define WIN_LD 3072         // W_in row stride (2*D_INNER)
#define NCH   16            // scan chunks
#define TCH   256           // timesteps per chunk

// GEMM1 (4096x768x1536) block tiling
#define BM    64
#define BN    128
#define KB    32
#define ALD   36            // LDS pitch for A panel rows (32 + 4 pad dwords)
#define BLD   132           // LDS pitch for B panel rows (128 + 4 pad dwords)
#define NPAN  (DM / KB)     // 24 K-panels

typedef float v2f __attribute__((ext_vector_type(2)));
typedef float v8f __attribute__((ext_vector_type(8)));
typedef unsigned int u32x4 __attribute__((ext_vector_type(4)));
typedef int i32x4 __attribute__((ext_vector_type(4)));
typedef int i32x8 __attribute__((ext_vector_type(8)));

// ---------------- Tensor Data Mover: async 2D tile -> LDS -------------------
// Builds a D# (ISA 8.3/8.4) for a 2D fp32 tile and issues TENSOR_LOAD_TO_LDS.
// dims/strides in elements; LDS row padding via pad codes (ISA group1 fields).
// This toolchain exposes the 6-arg builtin: (g0, g1, g2, g3, g4, cpol).
__device__ __forceinline__ void tdm_load_2d(unsigned lds_byte_off,
                                            const float* gptr,
                                            unsigned tile_w, unsigned tile_h,
                                            unsigned row_stride,
                                            unsigned pad_interval_code,
                                            unsigned pad_amount_code) {
  unsigned long long ga = (unsigned long long)(uintptr_t)gptr;
  u32x4 g0;
  g0[0] = 1u;                                   // count=1, user descriptor
  g0[1] = lds_byte_off;                         // lds_addr [63:32]
  g0[2] = (unsigned)(ga & 0xFFFFFFFFu);         // global_addr[31:0]
  g0[3] = (unsigned)((ga >> 32) & 0x01FFFFFFu)  // global_addr[56:32]
          | (2u << 30);                         // type = 2 ("image")
  i32x8 g1;
  unsigned w0 = (2u << 16)                      // data_size = 4 bytes
              | (1u << 20)                      // pad_enable
              | (pad_interval_code << 22)
              | (pad_amount_code << 25);
  g1[0] = (int)w0;
  g1[1] = (int)((tile_w & 0xFFFFu) << 16);                          // tensor_dim0 lo
  g1[2] = (int)(((tile_w >> 16) & 0xFFFFu) | ((tile_h & 0xFFFFu) << 16)); // dim0 hi | dim1 lo
  g1[3] = (int)(((tile_h >> 16) & 0xFFFFu) | ((tile_w & 0xFFFFu) << 16)); // dim1 hi | tile_dim0
  g1[4] = (int)(tile_h & 0xFFFFu);                                  // tile_dim1 | tile_dim2=0
  g1[5] = (int)row_stride;                                          // tensor_dim0_stride lo32
  g1[6] = 0;                                                        // stride0 hi | stride1 lo
  g1[7] = 0;                                                        // stride1 hi
  i32x4 z4 = {0, 0, 0, 0};                      // groups 2/3 unused (2D tile)
  i32x8 z8 = {0, 0, 0, 0, 0, 0, 0, 0};          // trailing group (6-arg form)
  __builtin_amdgcn_tensor_load_to_lds(g0, g1, z4, z4, z8, 0);
}

// ---------------- GEMM1: TDM-fed, LDS-staged, double-buffered WMMA ----------
// C[4096,1536] = A[4096,768] @ B(=W_in[:, :1536], row stride WIN_LD)
__global__ __launch_bounds__(256)
void k_gemm1_tdm(const float* __restrict__ A, const float* __restrict__ B,
                 float* __restrict__ C) {
  __shared__ float Ash[2][BM * ALD];   // 18 KB
  __shared__ float Bsh[2][KB * BLD];   // 33 KB
  const int lane = threadIdx.x & 31;
  const int wave = threadIdx.x >> 5;
  const int lm = lane & 15, half = lane >> 4;
  const int wm = wave & 3;             // 4 m-tiles of 16 -> 64 rows
  const int wn = wave >> 2;            // 2 n-groups of 64 cols
  const int m0 = (blockIdx.x % (L / BM)) * BM;
  const int n0 = (blockIdx.x / (L / BM)) * BN;

  // LDS byte offsets (ISA: flat LDS address low 32 bits == LDS offset)
  const unsigned aoff[2] = {(unsigned)(uintptr_t)&Ash[0][0],
                            (unsigned)(uintptr_t)&Ash[1][0]};
  const unsigned boff[2] = {(unsigned)(uintptr_t)&Bsh[0][0],
                            (unsigned)(uintptr_t)&Bsh[1][0]};

  v8f zero8 = {0.f, 0.f, 0.f, 0.f, 0.f, 0.f, 0.f, 0.f};
  v8f acc[4] = {zero8, zero8, zero8, zero8};

  if (wave == 0) {   // prologue: panel 0 in flight
    tdm_load_2d(aoff[0], A + (size_t)m0 * DM, KB, BM, DM, 4, 3);   // 32x64, pad every 32dw
    tdm_load_2d(boff[0], B + n0, BN, KB, WIN_LD, 6, 3);            // 128x32, pad every 128dw
  }

  for (int p = 0; p < NPAN; ++p) {
    const int buf = p & 1;
    if (wave == 0) {
      if (p + 1 < NPAN) {   // issue next panel, then wait for current (in-order)
        tdm_load_2d(aoff[(p + 1) & 1], A + (size_t)m0 * DM + (p + 1) * KB,
                    KB, BM, DM, 4, 3);
        tdm_load_2d(boff[(p + 1) & 1], B + (size_t)((p + 1) * KB) * WIN_LD + n0,
                    BN, KB, WIN_LD, 6, 3);
        __builtin_amdgcn_s_wait_tensorcnt(2);
      } else {
        __builtin_amdgcn_s_wait_tensorcnt(0);
      }
    }
    __syncthreads();                    // publish panel to all 8 waves

    const float* Ab = &Ash[buf][0];
    const float* Bb = &Bsh[buf][0];
    #pragma unroll
    for (int kk = 0; kk < KB; kk += 4) {
      // A 16x4 f32 fragment: lanes 0-15 hold (K,K+1), lanes 16-31 (K+2,K+3)
      v2f a = *(const v2f*)(Ab + (wm * 16 + lm) * ALD + kk + half * 2);
      #pragma unroll
      for (int j = 0; j < 4; ++j) {     // reuse A across 4 N-tiles
        const int col = wn * 64 + j * 16 + lm;
        v2f b;
        b.x = Bb[(kk + half) * BLD + col];
        b.y = Bb[(kk + 2 + half) * BLD + col];
        acc[j] = __builtin_amdgcn_wmma_f32_16x16x4_f32(false, a, false, b,
                                                       (short)0, acc[j],
                                                       false, false);
      }
    }
    __syncthreads();                    // done reading before next overwrite
  }

  #pragma unroll
  for (int j = 0; j < 4; ++j) {
    const int col = n0 + wn * 64 + j * 16 + lm;
    #pragma unroll
    for (int r = 0; r < 8; ++r)
      C[(size_t)(m0 + wm * 16 + half * 8 + r) * DI + col] = acc[j][r];
  }
}

// ---------------- direct fp32 WMMA tile (for the small GEMMs) ---------------
__device__ __forceinline__ v8f wmma_tile_f32(const float* __restrict__ A, int lda,
                                             const float* __restrict__ B, int ldb,
                                             int mBase, int nBase, int K, int lane) {
  const int lm   = lane & 15;
  const int half = lane >> 4;
  v8f acc = {0.f, 0.f, 0.f, 0.f, 0.f, 0.f, 0.f, 0.f};
  const float* arow = A + (size_t)(mBase + lm) * lda + half * 2;
  const float* bcol = B + nBase + lm;
  __builtin_prefetch(arow, 0, 0);       // -> global_prefetch_b8
  __builtin_prefetch(bcol, 0, 0);
  for (int k = 0; k < K; k += 4) {
    v2f a = *(const v2f*)(arow + k);
    v2f b;
    b.x = bcol[(size_t)(k + half) * ldb];
    b.y = bcol[(size_t)(k + 2 + half) * ldb];
    acc = __builtin_amdgcn_wmma_f32_16x16x4_f32(false, a, false, b,
                                                (short)0, acc, false, false);
  }
  return acc;
}

__global__ __launch_bounds__(256)
void k_gemm_wmma(const float* __restrict__ A, int lda,
                 const float* __restrict__ B, int ldb,
                 float* __restrict__ C, int ldc,
                 int M, int N, int K) {
  const int lane = threadIdx.x & 31;
  const int wave = threadIdx.x >> 5;
  const int tilesN = N >> 4;
  const int tile = blockIdx.x * 8 + wave;
  if (tile >= (M >> 4) * tilesN) return;
  const int mBase = (tile / tilesN) << 4;
  const int nBase = (tile % tilesN) << 4;
  v8f acc = wmma_tile_f32(A, lda, B, ldb, mBase, nBase, K, lane);
  const int lm = lane & 15, half = lane >> 4;
  #pragma unroll
  for (int j = 0; j < 8; ++j)
    C[(size_t)(mBase + half * 8 + j) * ldc + nBase + lm] = acc[j];
}

__global__ __launch_bounds__(256)
void k_gemm_wmma_softplus(const float* __restrict__ A, int lda,
                          const float* __restrict__ B, int ldb,
                          const float* __restrict__ bias,
                          float* __restrict__ C, int ldc,
                          int M, int N, int K) {
  const int lane = threadIdx.x & 31;
  const int wave = threadIdx.x >> 5;
  const int tilesN = N >> 4;
  const int tile = blockIdx.x * 8 + wave;
  if (tile >= (M >> 4) * tilesN) return;
  const int mBase = (tile / tilesN) << 4;
  const int nBase = (tile % tilesN) << 4;
  v8f acc = wmma_tile_f32(A, lda, B, ldb, mBase, nBase, K, lane);
  const int lm = lane & 15, half = lane >> 4;
  const float bv = bias[nBase + lm];
  #pragma unroll
  for (int j = 0; j < 8; ++j) {
    float v = acc[j] + bv;
    float sp = (v > 20.f) ? v : log1pf(__expf(v));   // softplus
    C[(size_t)(mBase + half * 8 + j) * ldc + nBase + lm] = sp;
  }
}

// ---------------- z_last = frame_embs[L-1] @ W_in[:, DI:2*DI] ----------------
__global__ __launch_bounds__(256)
void k_zlast(const float* __restrict__ E, const float* __restrict__ Win,
             float* __restrict__ zlast) {
  const int j = blockIdx.x * 256 + threadIdx.x;
  const float* e = E + (size_t)(L - 1) * DM;
  float s = 0.f;
  for (int k = 0; k < DM; ++k) s += e[k] * Win[(size_t)k * WIN_LD + DI + j];
  zlast[j] = s;
}

// ---------------- causal depthwise conv(4) + bias + silu --------------------
__global__ __launch_bounds__(256)
void k_conv_silu(const float* __restrict__ X0, const float* __restrict__ cw,
                 const float* __restrict__ cb, float* __restrict__ X) {
  const size_t i = (size_t)blockIdx.x * 256 + threadIdx.x;  // t*DI + d
  const int d = (int)(i % DI);
  const int t = (int)(i / DI);
  float acc = cb[d];
  #pragma unroll
  for (int k = 0; k < 4; ++k) {
    int ts = t + k - 3;
    if (ts >= 0) acc += X0[(size_t)ts * DI + d] * cw[d * 4 + k];
  }
  X[i] = acc / (1.f + __expf(-acc));   // silu
}

// ---------------- C_last[n] = X[L-1] @ W_xproj[:, 64+n] ---------------------
__global__ __launch_bounds__(256)
void k_clast(const float* __restrict__ X, const float* __restrict__ Wxp,
             float* __restrict__ Clast) {
  __shared__ float red[256];
  const int tid = threadIdx.x;
  const int n = tid & 15, s = tid >> 4;
  const float* xl = X + (size_t)(L - 1) * DI;
  float acc = 0.f;
  for (int d = s; d < DI; d += 16) acc += xl[d] * Wxp[(size_t)d * XC + RB + n];
  red[tid] = acc;
  __syncthreads();
  if (s == 0) {
    float tot = 0.f;
    for (int g = 0; g < 16; ++g) tot += red[g * 16 + n];
    Clast[n] = tot;
  }
}

// ---------------- scan stage 1: per-chunk sums of dt ------------------------
__global__ __launch_bounds__(256)
void k_chunk_sums(const float* __restrict__ DT, float* __restrict__ cs) {
  const int c = blockIdx.y;
  const int d = blockIdx.x * 256 + threadIdx.x;
  const float* p = DT + (size_t)c * TCH * DI + d;
  float s = 0.f;
  for (int t = 0; t < TCH; ++t) s += p[(size_t)t * DI];
  cs[c * DI + d] = s;
}

// ---------------- scan stage 2: suffix offsets across chunks ----------------
__global__ __launch_bounds__(256)
void k_suffix(const float* __restrict__ cs, float* __restrict__ Osuf) {
  const int d = blockIdx.x * 256 + threadIdx.x;
  float run = 0.f;
  for (int c = NCH - 1; c >= 0; --c) { Osuf[c * DI + d] = run; run += cs[c * DI + d]; }
}

// ---------------- scan stage 3: partial h per (chunk, d, n) -----------------
// h_final[d,n] = sum_t exp(A[d,n] * R[t,d]) * dt[t,d]*B[t,n]*x[t,d]
__global__ __launch_bounds__(256)
void k_scan_partial(const float* __restrict__ DT, const float* __restrict__ X,
                    const float* __restrict__ DBC, const float* __restrict__ A_log,
                    const float* __restrict__ Osuf, float* __restrict__ partial) {
  __shared__ float Bsh[TCH * DS];   // 16 KB: B[t,n] for this chunk
  const int c = blockIdx.y;
  const int d = blockIdx.x * 256 + threadIdx.x;
  const int t0 = c * TCH;
  for (int idx = threadIdx.x; idx < TCH * DS; idx += 256) {
    int t = idx >> 4, n = idx & 15;
    Bsh[idx] = DBC[(size_t)(t0 + t) * RB + RK + n];
  }
  __syncthreads();
  float Arow[DS], accn[DS];
  #pragma unroll
  for (int n = 0; n < DS; ++n) {
    Arow[n] = -__expf(A_log[(size_t)d * DS + n]);
    accn[n] = 0.f;
  }
  float r = Osuf[c * DI + d];       // suffix-dt strictly after this chunk
  for (int t = TCH - 1; t >= 0; --t) {
    const size_t off = (size_t)(t0 + t) * DI + d;
    const float dtv = DT[off];
    const float coef = dtv * X[off];
    #pragma unroll
    for (int n = 0; n < DS; ++n)
      accn[n] += __expf(Arow[n] * r) * (coef * Bsh[t * DS + n]);
    r += dtv;
  }
  float* pp = partial + ((size_t)c * DI + d) * DS;
  #pragma unroll
  for (int n = 0; n < DS; ++n) pp[n] = accn[n];
}

// ---------------- reduce chunks -> ys -> fused output gate ------------------
__global__ __launch_bounds__(256)
void k_reduce_y(const float* __restrict__ partial, const float* __restrict__ Clast,
                const float* __restrict__ X, const float* __restrict__ Dv,
                const float* __restrict__ zlast, float* __restrict__ yvec) {
  const int d = blockIdx.x * 256 + threadIdx.x;
  float h[DS];
  #pragma unroll
  for (int n = 0; n < DS; ++n) h[n] = 0.f;
  for (int c = 0; c < NCH; ++c) {
    const float* pp = partial + ((size_t)c * DI + d) * DS;
    #pragma unroll
    for (int n = 0; n < DS; ++n) h[n] += pp[n];
  }
  float ys = 0.f;
  #pragma unroll
  for (int n = 0; n < DS; ++n) ys += h[n] * Clast[n];
  const float xl = X[(size_t)(L - 1) * DI + d];
  const float zv = zlast[d];
  yvec[d] = (ys + xl * Dv[d]) * (zv / (1.f + __expf(-zv)));
}

// ---------------- out_vec = y @ W_out (GEMV) --------------------------------
__global__ __launch_bounds__(256)
void k_out_gemv(const float* __restrict__ yvec, const float* __restrict__ Wout,
                float* __restrict__ outvec) {
  const int m = blockIdx.x * 256 + threadIdx.x;
  float s = 0.f;
  for (int d = 0; d < DI; ++d) s += yvec[d] * Wout[(size_t)d * DM + m];
  outvec[m] = s;
}

// ---------------- L2-normalize the single output row ------------------------
__global__ __launch_bounds__(256)
void k_norm(const float* __restrict__ outvec, float* __restrict__ out) {
  __shared__ float red[256];
  __shared__ float nrm;
  const int tid = threadIdx.x;
  float s = 0.f;
  for (int m = tid; m < DM; m += 256) { float v = outvec[m]; s += v * v; }
  red[tid] = s;
  __syncthreads();
  for (int off = 128; off > 0; off >>= 1) {
    if (tid < off) red[tid] += red[tid + off];
    __syncthreads();
  }
  if (tid == 0) nrm = fmaxf(sqrtf(red[0]), 1e-12f);
  __syncthreads();
  for (int m = tid; m < DM; m += 256) out[m] = outvec[m] / nrm;
}

// ---------------------------------------------------------------------------
extern "C" void kernel_launch(void* const* d_in, const int* in_sizes, int n_in,
                              void* d_out, int out_size, void* d_ws, size_t ws_size,
                              hipStream_t stream) {
  const float* E     = (const float*)d_in[0];
  const float* Win   = (const float*)d_in[1];
  const float* convw = (const float*)d_in[2];
  const float* convb = (const float*)d_in[3];
  const float* Wxp   = (const float*)d_in[4];
  const float* Wdt   = (const float*)d_in[5];
  const float* bdt   = (const float*)d_in[6];
  const float* Alog  = (const float*)d_in[7];
  const float* Dv    = (const float*)d_in[8];
  const float* Wout  = (const float*)d_in[9];

  float* ws = (float*)d_ws;
  const size_t LD = (size_t)L * DI;
  float* X0      = ws;                 // L*DI (reused as DT after conv)
  float* X       = X0 + LD;
  float* DBC     = X + LD;             // L*RB
  float* partial = DBC + (size_t)L * RB;           // NCH*DI*DS
  float* cs      = partial + (size_t)NCH * DI * DS;
  float* Osuf    = cs + (size_t)NCH * DI;
  float* zlast   = Osuf + (size_t)NCH * DI;
  float* Clast   = zlast + DI;
  float* yvec    = Clast + DS;
  float* outvec  = yvec + DI;
  float* DT      = X0;

  // 1) X0 = frame_embs @ W_in[:, :DI]  (TDM-staged WMMA GEMM)
  k_gemm1_tdm<<<(L / BM) * (DI / BN), 256, 0, stream>>>(E, Win, X0);
  // 2) z_last
  k_zlast<<<DI / 256, 256, 0, stream>>>(E, Win, zlast);
  // 3) X = silu(conv(X0))
  k_conv_silu<<<(unsigned)(LD / 256), 256, 0, stream>>>(X0, convw, convb, X);
  // 4) DBC = X @ W_xproj[:, :64]
  k_gemm_wmma<<<(256 * (RB / 16)) / 8, 256, 0, stream>>>(X, DI, Wxp, XC,
                                                         DBC, RB, L, RB, DI);
  // 5) C_last
  k_clast<<<1, 256, 0, stream>>>(X, Wxp, Clast);
  // 6) DT = softplus(DBC[:, :48] @ W_dt + b_dt)
  k_gemm_wmma_softplus<<<(256 * (DI / 16)) / 8, 256, 0, stream>>>(
      DBC, RB, Wdt, DI, bdt, DT, DI, L, DI, RK);
  // 7-9) parallel suffix-sum scan
  k_chunk_sums<<<dim3(DI / 256, NCH), 256, 0, stream>>>(DT, cs);
  k_suffix<<<DI / 256, 256, 0, stream>>>(cs, Osuf);
  k_scan_partial<<<dim3(DI / 256, NCH), 256, 0, stream>>>(DT, X, DBC, Alog,
                                                          Osuf, partial);
  // 10-12) output path
  k_reduce_y<<<DI / 256, 256, 0, stream>>>(partial, Clast, X, Dv, zlast, yvec);
  k_out_gemv<<<DM / 256, 256, 0, stream>>>(yvec, Wout, outvec);
  k_norm<<<1, 256, 0, stream>>>(outvec, (float*)d_out);
}